// TemplatePairBlock_44375602102632
// MI455X (gfx1250) — compile-verified
//
#include <hip/hip_runtime.h>
#include <math.h>

#define N_ 256
#define T_ 2
#define H_ 4
#define NR_ (T_*N_*N_)   /* 131072 rows */

typedef float v2f __attribute__((ext_vector_type(2)));
typedef float v8f __attribute__((ext_vector_type(8)));

#define Z8 {0.f,0.f,0.f,0.f,0.f,0.f,0.f,0.f}

// f32 WMMA: D[16,16] += A[16,4] * B[4,16]  (V_WMMA_F32_16X16X4_F32)
__device__ __forceinline__ v8f wmma4(v2f a, v2f b, v8f c) {
  return __builtin_amdgcn_wmma_f32_16x16x4_f32(false, a, false, b, (short)0, c, false, false);
}

__device__ __forceinline__ int rowoff(int t, int i, int j, int flag) {
  return flag ? ((t*N_ + j)*N_ + i) : ((t*N_ + i)*N_ + j);
}
__device__ __forceinline__ float sigm(float x){ return 1.0f/(1.0f + __expf(-x)); }

// LayerNorm of a 64-float row into registers
__device__ __forceinline__ void ln64(const float* src, const float* w, const float* b, float* x){
  float mu = 0.f;
  #pragma unroll
  for (int c=0;c<64;++c){ x[c]=src[c]; mu+=x[c]; }
  mu *= (1.f/64.f);
  float var = 0.f;
  #pragma unroll
  for (int c=0;c<64;++c){ float d=x[c]-mu; var += d*d; }
  var *= (1.f/64.f);
  float rs = rsqrtf(var + 1e-5f);
  #pragma unroll
  for (int c=0;c<64;++c){ x[c] = (x[c]-mu)*rs*w[c] + b[c]; }
}

// ---------------- elementwise helpers ----------------
__global__ void k_fill(float* delta, const float* bo, int n){
  int i = blockIdx.x*blockDim.x + threadIdx.x;
  if (i < n) delta[i] = bo[i & 63];
}
__global__ void k_add(float* zc, const float* delta, int n){
  int i = blockIdx.x*blockDim.x + threadIdx.x;
  if (i < n) zc[i] += delta[i];
}

// ---------------- triangle bias: tb[t,h,j,k] ----------------
__global__ void k_tb(const float* z, const float* lnw, const float* lnb,
                     const float* wb, float* tb, int flag){
  int idx = blockIdx.x*blockDim.x + threadIdx.x;  // t*65536 + j*256 + k
  if (idx >= NR_) return;
  int k = idx & 255, j = (idx>>8)&255, t = idx>>16;
  float x[64];
  ln64(z + rowoff(t,j,k,flag)*64, lnw, lnb, x);
  #pragma unroll
  for (int h=0;h<4;++h){
    float s=0.f;
    #pragma unroll
    for (int c=0;c<64;++c) s += x[c]*wb[c*4+h];
    tb[((t*4+h)*N_+j)*N_+k] = s;
  }
}

// ---------------- attention head: one workgroup per (t,i,h) ----------------
__global__ void __launch_bounds__(128) k_att(
    const float* z, const float* mask,
    const float* lnw, const float* lnb,
    const float* wq, const float* wk, const float* wv,
    const float* wg, const float* bg, const float* wo,
    const float* tb, float* delta, int flag)
{
  // ShA serves two lives: phase 1-2 it is Xs[256][65] (LN'd rows);
  // phase 3 it is the 4 per-wave score strips Ss[4][16*257].
  __shared__ float ShA[256*65];
  __shared__ float Qh[N_*17];
  __shared__ float Kh[N_*17];
  __shared__ float Vh[N_*17];
  __shared__ float Gh[N_*17];

  const int bid = blockIdx.x;
  const int h = bid & 3;
  const int i = (bid >> 2) & 255;
  const int t = bid >> 10;
  const int tid = threadIdx.x;
  const int lane = tid & 31, wave = tid >> 5;
  const int l16 = lane & 15, half = lane >> 4;

  float* Xs = ShA;

  // phase 1: LN rows into LDS
  for (int j = tid; j < N_; j += 128) {
    float x[64];
    ln64(z + rowoff(t,i,j,flag)*64, lnw, lnb, x);
    #pragma unroll
    for (int c=0;c<64;++c) Xs[j*65+c] = x[c];
  }
  __syncthreads();

  // phase 2: head projections via WMMA (4 matrices x 16 row-tiles, K=64)
  for (int job = wave; job < 64; job += 4){
    int rt = job >> 2, mat = job & 3;
    const float* W = (mat==0)?wq:(mat==1)?wk:(mat==2)?wv:wg;
    v8f acc = Z8;
    #pragma unroll
    for (int kk=0; kk<64; kk+=4){
      int k0 = kk + 2*half;
      v2f a,b;
      a.x = Xs[(rt*16+l16)*65 + k0];  a.y = Xs[(rt*16+l16)*65 + k0+1];
      b.x = W[k0*64 + h*16 + l16];    b.y = W[(k0+1)*64 + h*16 + l16];
      acc = wmma4(a,b,acc);
    }
    float* DST = (mat==0)?Qh:(mat==1)?Kh:(mat==2)?Vh:Gh;
    if (mat == 3){
      #pragma unroll
      for (int r=0;r<8;++r)
        DST[(rt*16 + r + 8*half)*17 + l16] = sigm(acc[r] + bg[h*16 + l16]);
    } else {
      #pragma unroll
      for (int r=0;r<8;++r)
        DST[(rt*16 + r + 8*half)*17 + l16] = acc[r];
    }
  }
  __syncthreads();   // Xs dead; its LDS becomes the score strips

  float* S = ShA + wave*(16*257);
  for (int jt = wave; jt < 16; jt += 4) {
    // S = scale*Q.K^T + tb + INF*(mask-1)
    #pragma unroll 1
    for (int kt = 0; kt < 16; ++kt) {
      v8f acc = Z8;
      #pragma unroll
      for (int kk = 0; kk < 16; kk += 4) {
        int d0 = kk + 2*half;
        v2f a, b;
        a.x = Qh[(jt*16+l16)*17 + d0];  a.y = Qh[(jt*16+l16)*17 + d0 + 1];
        b.x = Kh[(kt*16+l16)*17 + d0];  b.y = Kh[(kt*16+l16)*17 + d0 + 1];
        acc = wmma4(a, b, acc);
      }
      #pragma unroll
      for (int r=0;r<8;++r){
        int jl = r + 8*half, kg = kt*16 + l16;
        int jg = jt*16 + jl;
        float m = flag ? mask[(t*N_+kg)*N_+i] : mask[(t*N_+i)*N_+kg];
        float bias = tb[((t*4+h)*N_+jg)*N_+kg];
        S[jl*257 + kg] = acc[r]*0.25f + bias + 100000.0f*(m-1.0f);
      }
    }
    // softmax over keys: lane handles (row=l16, half of columns), combine via shfl_xor 16
    {
      float* row = &S[l16*257 + half*128];
      float mx = -3.0e38f;
      #pragma unroll 1
      for (int c=0;c<128;++c) mx = fmaxf(mx, row[c]);
      mx = fmaxf(mx, __shfl_xor(mx, 16, 32));
      float sum = 0.f;
      #pragma unroll 1
      for (int c=0;c<128;++c){ float e=__expf(row[c]-mx); row[c]=e; sum+=e; }
      sum += __shfl_xor(sum, 16, 32);
      float inv = 1.f/sum;
      #pragma unroll 1
      for (int c=0;c<128;++c) row[c]*=inv;
    }
    // O = A @ V  [16j x 16d], K=256, two independent accumulator chains
    v8f o0 = Z8, o1 = Z8;
    #pragma unroll 1
    for (int kk = 0; kk < 256; kk += 8) {
      int k0 = kk + 2*half, k1 = kk + 4 + 2*half;
      v2f a,b;
      a.x = S[l16*257 + k0];        a.y = S[l16*257 + k0 + 1];
      b.x = Vh[k0*17 + l16];        b.y = Vh[(k0+1)*17 + l16];
      o0 = wmma4(a,b,o0);
      a.x = S[l16*257 + k1];        a.y = S[l16*257 + k1 + 1];
      b.x = Vh[k1*17 + l16];        b.y = Vh[(k1+1)*17 + l16];
      o1 = wmma4(a,b,o1);
    }
    // gate, stage Og into front of strip
    #pragma unroll
    for (int r=0;r<8;++r){
      int jl = r + 8*half;
      S[jl*257 + l16] = (o0[r] + o1[r]) * Gh[(jt*16+jl)*17 + l16];
    }
    // Y = Og @ Wo_h -> atomic accumulate into delta
    #pragma unroll 1
    for (int ct=0; ct<4; ++ct){
      v8f y = Z8;
      #pragma unroll
      for (int kk=0; kk<16; kk+=4){
        int d0 = kk + 2*half;
        v2f a,b;
        a.x = S[l16*257 + d0]; a.y = S[l16*257 + d0 + 1];
        b.x = wo[(h*16+d0)*64 + ct*16 + l16];
        b.y = wo[(h*16+d0+1)*64 + ct*16 + l16];
        y = wmma4(a,b,y);
      }
      #pragma unroll
      for (int r=0;r<8;++r){
        int jg = jt*16 + r + 8*half;
        atomicAdd(&delta[rowoff(t,i,jg,flag)*64 + ct*16 + l16], y[r]);
      }
    }
  }
}

// ---------------- triangle-mul precompute: ln, a, b ----------------
__global__ void __launch_bounds__(256) k_mulpre(
    const float* z, const float* mask,
    const float* lnw, const float* lnb,
    const float* wag, const float* bag, const float* wap, const float* bap,
    const float* wbg, const float* bbg, const float* wbp, const float* bbp,
    float* lnbuf, float* Ab, float* Bb)
{
  __shared__ float Xs[256*65];
  const int r0 = blockIdx.x * 256;
  const int tid = threadIdx.x;
  const int lane = tid & 31, wave = tid >> 5;
  const int l16 = lane & 15, half = lane >> 4;
  {
    int row = r0 + tid;
    float x[64];
    ln64(z + row*64, lnw, lnb, x);
    #pragma unroll
    for (int c=0;c<64;++c){ lnbuf[row*64+c]=x[c]; Xs[tid*65+c]=x[c]; }
  }
  __syncthreads();
  for (int job = wave; job < 64; job += 8){
    int rt = job >> 2, ct = job & 3;
    v8f g1=Z8, p1=Z8, g2=Z8, p2=Z8;
    #pragma unroll 1
    for (int kk=0; kk<64; kk+=4){
      int k0 = kk + 2*half;
      v2f a, b;
      a.x = Xs[(rt*16+l16)*65 + k0]; a.y = Xs[(rt*16+l16)*65 + k0+1];
      b.x=wag[k0*64+ct*16+l16]; b.y=wag[(k0+1)*64+ct*16+l16]; g1=wmma4(a,b,g1);
      b.x=wap[k0*64+ct*16+l16]; b.y=wap[(k0+1)*64+ct*16+l16]; p1=wmma4(a,b,p1);
      b.x=wbg[k0*64+ct*16+l16]; b.y=wbg[(k0+1)*64+ct*16+l16]; g2=wmma4(a,b,g2);
      b.x=wbp[k0*64+ct*16+l16]; b.y=wbp[(k0+1)*64+ct*16+l16]; p2=wmma4(a,b,p2);
    }
    #pragma unroll
    for (int r=0;r<8;++r){
      int rowg = r0 + rt*16 + r + 8*half;
      int c = ct*16 + l16;
      float m = mask[rowg];
      Ab[rowg*64+c] = m * sigm(g1[r]+bag[c]) * (p1[r]+bap[c]);
      Bb[rowg*64+c] = m * sigm(g2[r]+bbg[c]) * (p2[r]+bbp[c]);
    }
  }
}

// ---------------- triangle-mul main: x2 einsum + LN + gate + proj ----------------
__global__ void __launch_bounds__(256) k_mulmain(
    const float* Ab, const float* Bb, const float* lnbuf,
    const float* lnow, const float* lnob,
    const float* wgm, const float* bgm, const float* wz, const float* bz,
    float* zc, int outgoing)
{
  __shared__ float X2[256*65];   // x2 block, later its LN
  __shared__ float Ls[256*65];   // ln_in block (for the gate)
  const int b = blockIdx.x;
  const int jt = b & 15, it = (b>>4)&15, t = b>>8;
  const int tid = threadIdx.x;
  const int lane = tid&31, wave = tid>>5, l16=lane&15, half=lane>>4;

  // x2[i,j,c] = sum_k a(.)b(.) : one 256-deep WMMA matmul per channel
  for (int c = wave; c < 64; c += 8){
    const int ig = it*16 + l16, jg = jt*16 + l16;
    const size_t sA = outgoing ? (size_t)64 : (size_t)N_*64;
    const float* Abase = outgoing ? Ab + ((size_t)(t*N_+ig)*N_)*64 + c
                                  : Ab + ((size_t)t*N_*N_)*64 + (size_t)ig*64 + c;
    const float* Bbase = outgoing ? Bb + ((size_t)(t*N_+jg)*N_)*64 + c
                                  : Bb + ((size_t)t*N_*N_)*64 + (size_t)jg*64 + c;
    v8f acc0 = Z8, acc1 = Z8;
    #pragma unroll 1
    for (int kk=0; kk<256; kk+=8){
      int k0 = kk + 2*half, k1 = kk + 4 + 2*half;
      v2f a, bb;
      a.x  = Abase[(size_t)k0*sA];      a.y  = Abase[(size_t)(k0+1)*sA];
      bb.x = Bbase[(size_t)k0*sA];      bb.y = Bbase[(size_t)(k0+1)*sA];
      acc0 = wmma4(a, bb, acc0);
      a.x  = Abase[(size_t)k1*sA];      a.y  = Abase[(size_t)(k1+1)*sA];
      bb.x = Bbase[(size_t)k1*sA];      bb.y = Bbase[(size_t)(k1+1)*sA];
      acc1 = wmma4(a, bb, acc1);
      if (kk + 8 < 256) {
        __builtin_prefetch(Abase + (size_t)(kk+8)*sA, 0, 1);
        __builtin_prefetch(Bbase + (size_t)(kk+8)*sA, 0, 1);
      }
    }
    #pragma unroll
    for (int r=0;r<8;++r){
      int il = r + 8*half;
      X2[(il*16 + l16)*65 + c] = acc0[r] + acc1[r];
    }
  }
  __syncthreads();
  // per-(i,j) LayerNorm of x2 (back into LDS) + stage ln_in rows for the gate
  {
    int p = tid, il = p>>4, jl = p&15;
    float x2[64];
    ln64(&X2[p*65], lnow, lnob, x2);
    int rowg = (t*N_ + it*16+il)*N_ + jt*16 + jl;
    #pragma unroll
    for (int c=0;c<64;++c){ X2[p*65+c] = x2[c]; Ls[p*65+c] = lnbuf[rowg*64+c]; }
  }
  __syncthreads();
  // out = (x2ln @ wz + bz) * sigmoid(ln_in @ wg + bg)  -- both GEMMs on WMMA
  for (int job = wave; job < 64; job += 8){
    int rt = job >> 2, ct = job & 3;
    v8f ga = Z8, za = Z8;
    #pragma unroll 1
    for (int kk=0; kk<64; kk+=4){
      int k0 = kk + 2*half;
      v2f a, b;
      a.x = Ls[(rt*16+l16)*65 + k0]; a.y = Ls[(rt*16+l16)*65 + k0+1];
      b.x = wgm[k0*64 + ct*16 + l16]; b.y = wgm[(k0+1)*64 + ct*16 + l16];
      ga = wmma4(a,b,ga);
      a.x = X2[(rt*16+l16)*65 + k0]; a.y = X2[(rt*16+l16)*65 + k0+1];
      b.x = wz[k0*64 + ct*16 + l16]; b.y = wz[(k0+1)*64 + ct*16 + l16];
      za = wmma4(a,b,za);
    }
    #pragma unroll
    for (int r=0;r<8;++r){
      int p2 = rt*16 + r + 8*half;
      int rowg = (t*N_ + it*16 + (p2>>4))*N_ + jt*16 + (p2&15);
      int c = ct*16 + l16;
      zc[rowg*64+c] += (za[r] + bz[c]) * sigm(ga[r] + bgm[c]);
    }
  }
}

// ---------------- transition MLP ----------------
__global__ void __launch_bounds__(256) k_trans(
    const float* lnw, const float* lnb,
    const float* w1, const float* b1, const float* w2, const float* b2,
    float* zc)
{
  __shared__ float Xs[256*65];
  __shared__ float Hs[8][16*132];
  const int r0 = blockIdx.x*256;
  const int tid = threadIdx.x;
  const int lane=tid&31, wave=tid>>5, l16=lane&15, half=lane>>4;
  {
    int row = r0 + tid;
    float x[64];
    ln64(zc + row*64, lnw, lnb, x);
    #pragma unroll
    for (int c=0;c<64;++c) Xs[tid*65+c]=x[c];
  }
  __syncthreads();
  float* Hw = &Hs[wave][0];
  for (int rt = wave; rt < 16; rt += 8){
    #pragma unroll 1
    for (int c2=0;c2<8;++c2){
      v8f acc = Z8;
      #pragma unroll
      for (int kk=0;kk<64;kk+=4){
        int k0=kk+2*half;
        v2f a,b;
        a.x = Xs[(rt*16+l16)*65+k0]; a.y = Xs[(rt*16+l16)*65+k0+1];
        b.x = w1[k0*128 + c2*16+l16]; b.y = w1[(k0+1)*128 + c2*16+l16];
        acc = wmma4(a,b,acc);
      }
      #pragma unroll
      for (int r=0;r<8;++r){
        int rl = r+8*half, cl = c2*16+l16;
        Hw[rl*132 + cl] = fmaxf(acc[r] + b1[cl], 0.f);
      }
    }
    #pragma unroll 1
    for (int ct=0;ct<4;++ct){
      v8f acc0 = Z8, acc1 = Z8;
      #pragma unroll 1
      for (int kk=0;kk<128;kk+=8){
        int k0=kk+2*half, k1=kk+4+2*half;
        v2f a,b;
        a.x = Hw[l16*132 + k0]; a.y = Hw[l16*132 + k0+1];
        b.x = w2[k0*64 + ct*16+l16]; b.y = w2[(k0+1)*64 + ct*16+l16];
        acc0 = wmma4(a,b,acc0);
        a.x = Hw[l16*132 + k1]; a.y = Hw[l16*132 + k1+1];
        b.x = w2[k1*64 + ct*16+l16]; b.y = w2[(k1+1)*64 + ct*16+l16];
        acc1 = wmma4(a,b,acc1);
      }
      #pragma unroll
      for (int r=0;r<8;++r){
        int rowg = r0 + rt*16 + r + 8*half;
        int c = ct*16+l16;
        zc[rowg*64+c] += acc0[r] + acc1[r] + b2[c];
      }
    }
  }
}

// ---------------- launcher ----------------
// input order (setup_inputs dict order, nested dicts in insertion order):
// 0:z 1:mask
// att_start 2..11:  ln_w ln_b wq wk wv wb wg bg wo bo
// att_end  12..21:  (same)
// mul_out  22..37:  ln_in_w ln_in_b ln_out_w ln_out_b wag bag wap bap wbg bbg wbp bbp wg bg wz bz
// mul_in   38..53:  (same)
// trans    54..59:  ln_w ln_b w1 b1 w2 b2
extern "C" void kernel_launch(void* const* d_in, const int* in_sizes, int n_in,
                              void* d_out, int out_size, void* d_ws, size_t ws_size,
                              hipStream_t stream) {
  (void)in_sizes; (void)n_in; (void)out_size; (void)ws_size;
  const float* z    = (const float*)d_in[0];
  const float* mask = (const float*)d_in[1];
  const float* const* P = (const float* const*)d_in;

  float* zc    = (float*)d_out;
  float* lnbuf = (float*)d_ws;                      // 33.5 MB
  float* Ab    = lnbuf + (size_t)NR_*64;            // 33.5 MB
  float* Bb    = Ab    + (size_t)NR_*64;            // 33.5 MB
  float* tbb   = Bb    + (size_t)NR_*64;            // 2.1 MB
  float* delta = Ab;                                // alias (phases disjoint)

  const int NE = NR_*64;
  hipMemcpyAsync(zc, z, (size_t)NE*sizeof(float), hipMemcpyDeviceToDevice, stream);

  // two attention passes: row-wise (flag=0), column-wise transposed (flag=1)
  for (int pass = 0; pass < 2; ++pass) {
    int base = 2 + pass*10;
    int flag = pass;
    k_fill<<<NE/256, 256, 0, stream>>>(delta, P[base+9], NE);
    k_tb  <<<NR_/256, 256, 0, stream>>>(zc, P[base+0], P[base+1], P[base+5], tbb, flag);
    k_att <<<T_*N_*H_, 128, 0, stream>>>(zc, mask, P[base+0], P[base+1],
                                         P[base+2], P[base+3], P[base+4],
                                         P[base+6], P[base+7], P[base+8],
                                         tbb, delta, flag);
    k_add <<<NE/256, 256, 0, stream>>>(zc, delta, NE);
  }

  // two triangle-mul passes: outgoing then incoming
  for (int pass = 0; pass < 2; ++pass) {
    int base = 22 + pass*16;
    int outgoing = (pass == 0) ? 1 : 0;
    k_mulpre <<<NR_/256, 256, 0, stream>>>(zc, mask, P[base+0], P[base+1],
                                           P[base+4], P[base+5], P[base+6], P[base+7],
                                           P[base+8], P[base+9], P[base+10], P[base+11],
                                           lnbuf, Ab, Bb);
    k_mulmain<<<T_*16*16, 256, 0, stream>>>(Ab, Bb, lnbuf, P[base+2], P[base+3],
                                            P[base+12], P[base+13], P[base+14], P[base+15],
                                            zc, outgoing);
  }

  // transition
  k_trans<<<NR_/256, 256, 0, stream>>>(P[54], P[55], P[56], P[57], P[58], P[59], zc);
}